// ROIAlign_13615046329082
// MI455X (gfx1250) — compile-verified
//
#include <hip/hip_runtime.h>
#include <hip/hip_bf16.h>
#include <stdint.h>

#define B_      2
#define C_      256
#define H_      200
#define W_      304
#define OUT_H_  7
#define OUT_W_  7
#define NPIX    49      // 7*7
#define NTAP    16      // 4 bilinear samples * 4 corners
#define SCALE_  0.25f
#define CSPLIT  4       // channel slices per ROI -> 64 channels per block
#define CPERB   (C_ / CSPLIT)

__device__ __forceinline__ void corners1d(float v, int size,
                                          int& i0, int& i1,
                                          float& w0, float& w1, bool& valid) {
    valid = (v >= -1.0f) && (v <= (float)size);
    float vc = fminf(fmaxf(v, 0.0f), (float)(size - 1));
    float fl = floorf(vc);
    i0 = (int)fl;
    i1 = min(i0 + 1, size - 1);
    float l = vc - fl;
    w0 = 1.0f - l;
    w1 = l;
}

__global__ __launch_bounds__(256) void roi_align_gfx1250(
    const float* __restrict__ feat,   // (B, C, H, W)
    const float* __restrict__ rois,   // (K, 5)
    float* __restrict__ out)          // (K, C, 7, 7)
{
    __shared__ float s_roi[8];
    __shared__ int   s_idx[NTAP * NPIX];
    __shared__ float s_w  [NTAP * NPIX];

    const int k   = blockIdx.x;
    const int cb  = blockIdx.y;        // channel slice 0..3
    const int tid = threadIdx.x;

    // ---- Phase A0: async-copy this ROI's 5 floats global -> LDS (ASYNCcnt) ----
    if (tid < 5) {
        // low 32 bits of a generic pointer to __shared__ == LDS byte offset
        uint32_t lds_off = (uint32_t)(uintptr_t)(&s_roi[tid]);
        uint64_t gaddr   = (uint64_t)(uintptr_t)(rois + (size_t)k * 5 + tid);
        asm volatile("global_load_async_to_lds_b32 %0, %1, off"
                     :: "v"(lds_off), "v"(gaddr) : "memory");
    }
    asm volatile("s_wait_asynccnt 0" ::: "memory");
    __syncthreads();

    const int   bidx  = (int)s_roi[0];
    const float sx    = s_roi[1] * SCALE_;
    const float sy    = s_roi[2] * SCALE_;
    const float ex    = s_roi[3] * SCALE_;
    const float ey    = s_roi[4] * SCALE_;
    const float roi_w = fmaxf(ex - sx, 1.0f);
    const float roi_h = fmaxf(ey - sy, 1.0f);
    const float bin_w = roi_w * (1.0f / OUT_W_);
    const float bin_h = roi_h * (1.0f / OUT_H_);

    // ---- Phase A1: build per-ROI tap tables (196 sample points -> 784 taps) ----
    if (tid < NPIX * 4) {
        const int p    = tid >> 2;        // pixel 0..48
        const int s    = tid & 3;         // sample 0..3
        const int ph   = p / OUT_W_;
        const int pw   = p % OUT_W_;
        const int sy_i = s >> 1;
        const int sx_i = s & 1;
        const float fy = 0.25f + 0.5f * (float)sy_i;   // (i+0.5)/SAMP
        const float fx = 0.25f + 0.5f * (float)sx_i;
        const float y  = sy + ((float)ph + fy) * bin_h;
        const float x  = sx + ((float)pw + fx) * bin_w;

        int iy0, iy1, ix0, ix1;
        float wy0, wy1, wx0, wx1;
        bool vy, vx;
        corners1d(y, H_, iy0, iy1, wy0, wy1, vy);
        corners1d(x, W_, ix0, ix1, wx0, wx1, vx);
        const float vmul = (vy && vx) ? 1.0f : 0.0f;

        const int t0 = s * 4;
        s_idx[(t0 + 0) * NPIX + p] = iy0 * W_ + ix0;
        s_idx[(t0 + 1) * NPIX + p] = iy0 * W_ + ix1;
        s_idx[(t0 + 2) * NPIX + p] = iy1 * W_ + ix0;
        s_idx[(t0 + 3) * NPIX + p] = iy1 * W_ + ix1;
        s_w  [(t0 + 0) * NPIX + p] = wy0 * wx0 * vmul;
        s_w  [(t0 + 1) * NPIX + p] = wy0 * wx1 * vmul;
        s_w  [(t0 + 2) * NPIX + p] = wy1 * wx0 * vmul;
        s_w  [(t0 + 3) * NPIX + p] = wy1 * wx1 * vmul;
    }
    __syncthreads();

    // ---- Phase B: gather + weighted sum. 245 threads = 49 pixels x 5 channel-lanes ----
    if (tid < NPIX * 5) {
        const int p  = tid % NPIX;   // consecutive lanes -> consecutive pixels (line sharing)
        const int c0 = tid / NPIX;   // 0..4, stride-5 channel walk covers all residues

        // taps are loop-invariant over channels: hoist to registers
        int   idxs[NTAP];
        float ws  [NTAP];
#pragma unroll
        for (int t = 0; t < NTAP; ++t) {
            idxs[t] = s_idx[t * NPIX + p];
            ws  [t] = s_w  [t * NPIX + p];
        }

        const size_t HW      = (size_t)(H_ * W_);
        const float* fplane0 = feat + (size_t)bidx * C_ * HW;

        for (int cl = c0; cl < CPERB; cl += 5) {
            const int    c  = cb * CPERB + cl;
            const float* fb = fplane0 + (size_t)c * HW;

            // prefetch next channel plane's first gather line (L2-resident working set)
            if (cl + 5 < CPERB)
                __builtin_prefetch(fb + 5 * HW + idxs[0], 0, 3);

            float acc = 0.0f;
#pragma unroll
            for (int t = 0; t < NTAP; ++t)
                acc = fmaf(ws[t], fb[idxs[t]], acc);

            // write-once output: non-temporal store, keep L2 for the feature map
            __builtin_nontemporal_store(acc * 0.25f,
                                        out + ((size_t)k * C_ + c) * NPIX + p);
        }
    }
}

extern "C" void kernel_launch(void* const* d_in, const int* in_sizes, int n_in,
                              void* d_out, int out_size, void* d_ws, size_t ws_size,
                              hipStream_t stream) {
    const float* feat = (const float*)d_in[0];
    const float* rois = (const float*)d_in[1];
    float*       out  = (float*)d_out;
    (void)d_ws; (void)ws_size; (void)out_size; (void)n_in;

    const int K = in_sizes[1] / 5;   // 512
    dim3 grid(K, CSPLIT);
    roi_align_gfx1250<<<grid, 256, 0, stream>>>(feat, rois, out);
}